// WaveletAttention_77120432767309
// MI455X (gfx1250) — compile-verified
//
#include <hip/hip_runtime.h>
#include <hip/hip_bf16.h>

typedef __attribute__((ext_vector_type(2))) float v2f;
typedef __attribute__((ext_vector_type(8))) float v8f;

// ---------------- workspace layout (float offsets) ----------------
static constexpr size_t SZ_L3   = (size_t)128 * 4096;     // 64x64 per image
static constexpr size_t SZ_L2   = (size_t)128 * 16384;    // 128x128 per image
static constexpr size_t SZ_L1   = (size_t)128 * 65536;    // 256x256 per image
static constexpr size_t OFF_A3  = 0;
static constexpr size_t OFF_CH3 = SZ_L3 * 1;
static constexpr size_t OFF_CV3 = SZ_L3 * 2;
static constexpr size_t OFF_CD3 = SZ_L3 * 3;
static constexpr size_t OFF_CH2 = SZ_L3 * 4;
static constexpr size_t OFF_CV2 = OFF_CH2 + SZ_L2;
static constexpr size_t OFF_CD2 = OFF_CH2 + 2 * SZ_L2;
static constexpr size_t OFF_CH1 = OFF_CH2 + 3 * SZ_L2;
static constexpr size_t OFF_CV1 = OFF_CH1 + SZ_L1;
static constexpr size_t OFF_CD1 = OFF_CH1 + 2 * SZ_L1;
static constexpr size_t OFF_PART  = OFF_CH1 + 3 * SZ_L1;  // 512 blocks * 384 floats
static constexpr size_t OFF_LOGIT = OFF_PART + (size_t)512 * 384; // 1280 floats
static constexpr size_t OFF_GATE  = OFF_LOGIT + 1280;             // 1280 floats

// =================================================================
// Kernel 1: fused 3-level Haar DWT. One thread = one 8x8 block of
// 4 channels (channels innermost -> float4 loads).
// =================================================================
__global__ __launch_bounds__(256) void dwt_kernel(const float* __restrict__ in,
                                                  float* __restrict__ ws) {
    unsigned g = blockIdx.x * 256u + threadIdx.x;   // 131072 threads
    int c0 = (g & 1) * 4;
    int j3 = (g >> 1) & 63;
    int i3 = (g >> 7) & 63;
    int b  = g >> 13;

    float a2[4][2][2];
#pragma unroll
    for (int u = 0; u < 2; ++u) {
#pragma unroll
        for (int v = 0; v < 2; ++v) {
            float a1[4][2][2];
#pragma unroll
            for (int pp = 0; pp < 2; ++pp) {
#pragma unroll
                for (int qq = 0; qq < 2; ++qq) {
                    int y = 8 * i3 + 4 * u + 2 * pp;
                    int x = 8 * j3 + 4 * v + 2 * qq;
                    size_t p00 = ((size_t)(b * 512 + y) * 512 + x) * 8 + c0;
                    float4 v00 = *(const float4*)(in + p00);
                    float4 v01 = *(const float4*)(in + p00 + 8);
                    float4 v10 = *(const float4*)(in + p00 + 4096);
                    float4 v11 = *(const float4*)(in + p00 + 4104);
                    float q00[4] = {v00.x, v00.y, v00.z, v00.w};
                    float q01[4] = {v01.x, v01.y, v01.z, v01.w};
                    float q10[4] = {v10.x, v10.y, v10.z, v10.w};
                    float q11[4] = {v11.x, v11.y, v11.z, v11.w};
                    int l1r = 4 * i3 + 2 * u + pp;
                    int l1c = 4 * j3 + 2 * v + qq;
#pragma unroll
                    for (int k = 0; k < 4; ++k) {
                        float s  = q00[k] + q01[k] + q10[k] + q11[k];
                        float h  = q00[k] + q01[k] - q10[k] - q11[k]; // H-detail
                        float vv = q00[k] - q01[k] + q10[k] - q11[k]; // W-detail
                        float d  = q00[k] - q01[k] - q10[k] + q11[k];
                        a1[k][pp][qq] = 0.5f * s;
                        int n = b * 8 + c0 + k;
                        size_t l1i = (size_t)n * 65536 + (size_t)l1r * 256 + l1c;
                        ws[OFF_CH1 + l1i] = 0.5f * h;
                        ws[OFF_CV1 + l1i] = 0.5f * vv;
                        ws[OFF_CD1 + l1i] = 0.5f * d;
                    }
                }
            }
            int l2r = 2 * i3 + u, l2c = 2 * j3 + v;
#pragma unroll
            for (int k = 0; k < 4; ++k) {
                float s  = a1[k][0][0] + a1[k][0][1] + a1[k][1][0] + a1[k][1][1];
                float h  = a1[k][0][0] + a1[k][0][1] - a1[k][1][0] - a1[k][1][1];
                float vv = a1[k][0][0] - a1[k][0][1] + a1[k][1][0] - a1[k][1][1];
                float d  = a1[k][0][0] - a1[k][0][1] - a1[k][1][0] + a1[k][1][1];
                a2[k][u][v] = 0.5f * s;
                int n = b * 8 + c0 + k;
                size_t l2i = (size_t)n * 16384 + (size_t)l2r * 128 + l2c;
                ws[OFF_CH2 + l2i] = 0.5f * h;
                ws[OFF_CV2 + l2i] = 0.5f * vv;
                ws[OFF_CD2 + l2i] = 0.5f * d;
            }
        }
    }
#pragma unroll
    for (int k = 0; k < 4; ++k) {
        float s  = a2[k][0][0] + a2[k][0][1] + a2[k][1][0] + a2[k][1][1];
        float h  = a2[k][0][0] + a2[k][0][1] - a2[k][1][0] - a2[k][1][1];
        float vv = a2[k][0][0] - a2[k][0][1] + a2[k][1][0] - a2[k][1][1];
        float d  = a2[k][0][0] - a2[k][0][1] - a2[k][1][0] + a2[k][1][1];
        int n = b * 8 + c0 + k;
        size_t l3i = (size_t)n * 4096 + (size_t)i3 * 64 + j3;
        ws[OFF_A3  + l3i] = 0.5f * s;
        ws[OFF_CH3 + l3i] = 0.5f * h;
        ws[OFF_CV3 + l3i] = 0.5f * vv;
        ws[OFF_CD3 + l3i] = 0.5f * d;
    }
}

// =================================================================
// Kernel 2: gate-logit GEMM  D[128, LDW] += Coeff[128,K] x W[K,LDW]
// via V_WMMA_F32_16X16X4_F32. Templated on LDW/KPLANE so all index
// math constant-folds. One block = one (plane, 512-wide K chunk);
// 8 waves cover 128 images (16 each). W chunk staged transposed in
// LDS so each lane's B fragment is a single aligned ds_load_b64.
// Lanes with li >= LDW use a clamped column (duplicate compute, no
// divergence in the hot loop; unused columns are never stored).
// Partial D tiles go to workspace for a deterministic reduction.
// =================================================================
template <int LDW, int KPLANE>
__global__ __launch_bounds__(256) void logit_gemm_kernel(
    const float* __restrict__ W, float* __restrict__ ws,
    size_t planeBase0, size_t planeSize, int bidOffset) {
    constexpr int CHUNKS = KPLANE / 512;
    __shared__ float BsT[LDW][512];

    int t = blockIdx.x / CHUNKS;          // plane (cH/cV/cD), 0 for CA
    int c = blockIdx.x % CHUNKS;          // K chunk
    size_t planeBase = planeBase0 + (size_t)t * planeSize;
    size_t wRow0 = (size_t)t * KPLANE + (size_t)c * 512;
    int k0 = c * 512;

    // stage W[wRow0 .. wRow0+512)[0..LDW) transposed into LDS
    for (int i = threadIdx.x; i < 512 * LDW; i += 256) {
        int r = i / LDW, col = i % LDW;   // contiguous global read
        BsT[col][r] = W[wRow0 * LDW + i];
    }
    __syncthreads();

    int lane = threadIdx.x & 31;
    int wv   = threadIdx.x >> 5;          // M-tile: images [16*wv, 16*wv+16)
    int half = lane >> 4;                 // 0: K={0,1}; 1: K={2,3}
    int li   = lane & 15;                 // A-row / B-col within tile
    int colB = (li < LDW) ? li : (LDW - 1);

    const float* Arow = ws + planeBase + (size_t)(wv * 16 + li) * KPLANE + k0;
    const float* Brow = &BsT[colB][2 * half];
    v8f acc = {0.f, 0.f, 0.f, 0.f, 0.f, 0.f, 0.f, 0.f};

#pragma unroll 8
    for (int kk = 0; kk < 512; kk += 4) {
        float2 av = *(const float2*)(Arow + kk + 2 * half); // global, 8B
        float2 bv = *(const float2*)(Brow + kk);            // LDS, 8B
        v2f a;   a.x = av.x;   a.y = av.y;
        v2f bfr; bfr.x = bv.x; bfr.y = bv.y;
        acc = __builtin_amdgcn_wmma_f32_16x16x4_f32(
            /*neg_a=*/false, a, /*neg_b=*/false, bfr,
            /*c_mod=*/(short)0, acc, /*reuse_a=*/false, /*reuse_b=*/false);
    }

    if (li < LDW) {
        float* part = ws + OFF_PART + (size_t)(bidOffset + blockIdx.x) * 384;
#pragma unroll
        for (int r = 0; r < 8; ++r) {
            int img = wv * 16 + half * 8 + r;   // D row r of this half
            part[img * LDW + li] = acc[r];
        }
    }
}

// =================================================================
// Kernel 2b: deterministic fixed-order reduction of GEMM partials.
// =================================================================
__global__ void logit_reduce_kernel(float* __restrict__ ws) {
    int t = blockIdx.x * 256 + threadIdx.x;
    if (t >= 1280) return;
    int b0, b1, idx;
    if (t < 128)      { b0 = 0;   b1 = 8;   idx = t;       }  // dotCA
    else if (t < 512) { b0 = 8;   b1 = 32;  idx = t - 128; }  // dotD3
    else if (t < 896) { b0 = 32;  b1 = 128; idx = t - 512; }  // dotD2
    else              { b0 = 128; b1 = 512; idx = t - 896; }  // dotD1
    float s = 0.f;
    for (int bb = b0; bb < b1; ++bb)
        s += ws[OFF_PART + (size_t)bb * 384 + idx];
    ws[OFF_LOGIT + t] = s;
}

// =================================================================
// Kernel 3: gates = sigmoid / softmax of logits (+bias). 128 threads.
// =================================================================
__global__ void gate_kernel(const float* __restrict__ bca,
                            const float* __restrict__ bd3,
                            const float* __restrict__ bd2,
                            const float* __restrict__ bd1,
                            float* __restrict__ ws) {
    int img = threadIdx.x;
    if (img >= 128) return;
    const float* L = ws + OFF_LOGIT;
    float* G = ws + OFF_GATE;
    G[img] = 1.f / (1.f + __expf(-(L[img] + bca[0])));
    const float* bs[3] = {bd3, bd2, bd1};
#pragma unroll
    for (int lev = 0; lev < 3; ++lev) {
        int base = 128 + lev * 384;
        float l0 = L[base + img * 3 + 0] + bs[lev][0];
        float l1 = L[base + img * 3 + 1] + bs[lev][1];
        float l2 = L[base + img * 3 + 2] + bs[lev][2];
        float m = fmaxf(l0, fmaxf(l1, l2));
        float e0 = __expf(l0 - m), e1 = __expf(l1 - m), e2 = __expf(l2 - m);
        float inv = 1.f / (e0 + e1 + e2);
        G[base + img * 3 + 0] = e0 * inv;
        G[base + img * 3 + 1] = e1 * inv;
        G[base + img * 3 + 2] = e2 * inv;
    }
}

// =================================================================
// Kernel 4: fused gated 3-level inverse Haar DWT.
// One thread = one 8x8 output block of 4 channels.
// =================================================================
__global__ __launch_bounds__(256) void idwt_kernel(const float* __restrict__ ws,
                                                   float* __restrict__ out) {
    unsigned g = blockIdx.x * 256u + threadIdx.x;
    int c0 = (g & 1) * 4;
    int j3 = (g >> 1) & 63;
    int i3 = (g >> 7) & 63;
    int b  = g >> 13;

    const float* G = ws + OFF_GATE;
    float w2g[4][3], w1g[4][3];
    float A3v[4], H3[4], V3[4], D3[4];
#pragma unroll
    for (int k = 0; k < 4; ++k) {
        int n = b * 8 + c0 + k;
        float aw = G[n];
        float w30 = G[128 + n * 3 + 0], w31 = G[128 + n * 3 + 1], w32 = G[128 + n * 3 + 2];
#pragma unroll
        for (int j = 0; j < 3; ++j) {
            w2g[k][j] = G[512 + n * 3 + j];
            w1g[k][j] = G[896 + n * 3 + j];
        }
        size_t l3i = (size_t)n * 4096 + (size_t)i3 * 64 + j3;
        A3v[k] = aw  * ws[OFF_A3  + l3i];
        H3[k]  = w30 * ws[OFF_CH3 + l3i];
        V3[k]  = w31 * ws[OFF_CV3 + l3i];
        D3[k]  = w32 * ws[OFF_CD3 + l3i];
    }

#pragma unroll
    for (int u = 0; u < 2; ++u) {
#pragma unroll
        for (int v = 0; v < 2; ++v) {
            float su = u ? -1.f : 1.f, sv = v ? -1.f : 1.f;
            float R2[4], C2h[4], C2v[4], C2d[4];
            int l2r = 2 * i3 + u, l2c = 2 * j3 + v;
#pragma unroll
            for (int k = 0; k < 4; ++k) {
                R2[k] = 0.5f * (A3v[k] + su * H3[k] + sv * V3[k] + su * sv * D3[k]);
                int n = b * 8 + c0 + k;
                size_t l2i = (size_t)n * 16384 + (size_t)l2r * 128 + l2c;
                C2h[k] = w2g[k][0] * ws[OFF_CH2 + l2i];
                C2v[k] = w2g[k][1] * ws[OFF_CV2 + l2i];
                C2d[k] = w2g[k][2] * ws[OFF_CD2 + l2i];
            }
#pragma unroll
            for (int pp = 0; pp < 2; ++pp) {
#pragma unroll
                for (int qq = 0; qq < 2; ++qq) {
                    float sp = pp ? -1.f : 1.f, sq = qq ? -1.f : 1.f;
                    float R1[4], C1h[4], C1v[4], C1d[4];
                    int l1r = 4 * i3 + 2 * u + pp;
                    int l1c = 4 * j3 + 2 * v + qq;
#pragma unroll
                    for (int k = 0; k < 4; ++k) {
                        R1[k] = 0.5f * (R2[k] + sp * C2h[k] + sq * C2v[k] + sp * sq * C2d[k]);
                        int n = b * 8 + c0 + k;
                        size_t l1i = (size_t)n * 65536 + (size_t)l1r * 256 + l1c;
                        C1h[k] = w1g[k][0] * ws[OFF_CH1 + l1i];
                        C1v[k] = w1g[k][1] * ws[OFF_CV1 + l1i];
                        C1d[k] = w1g[k][2] * ws[OFF_CD1 + l1i];
                    }
#pragma unroll
                    for (int rr = 0; rr < 2; ++rr) {
#pragma unroll
                        for (int cc = 0; cc < 2; ++cc) {
                            float sr = rr ? -1.f : 1.f, sc = cc ? -1.f : 1.f;
                            float4 o;
                            float* op = (float*)&o;
#pragma unroll
                            for (int k = 0; k < 4; ++k)
                                op[k] = 0.5f * (R1[k] + sr * C1h[k] + sc * C1v[k] +
                                                sr * sc * C1d[k]);
                            int y = 8 * i3 + 4 * u + 2 * pp + rr;
                            int x = 8 * j3 + 4 * v + 2 * qq + cc;
                            *(float4*)(out + ((size_t)(b * 512 + y) * 512 + x) * 8 + c0) = o;
                        }
                    }
                }
            }
        }
    }
}

// =================================================================
extern "C" void kernel_launch(void* const* d_in, const int* in_sizes, int n_in,
                              void* d_out, int out_size, void* d_ws, size_t ws_size,
                              hipStream_t stream) {
    const float* inp = (const float*)d_in[0];
    const float* Wca = (const float*)d_in[1];
    const float* bca = (const float*)d_in[2];
    const float* Wd3 = (const float*)d_in[3];
    const float* bd3 = (const float*)d_in[4];
    const float* Wd2 = (const float*)d_in[5];
    const float* bd2 = (const float*)d_in[6];
    const float* Wd1 = (const float*)d_in[7];
    const float* bd1 = (const float*)d_in[8];
    float* ws  = (float*)d_ws;
    float* out = (float*)d_out;

    dwt_kernel<<<512, 256, 0, stream>>>(inp, ws);

    logit_gemm_kernel<1, 4096>  <<<8,   256, 0, stream>>>(Wca, ws, OFF_A3,  SZ_L3, 0);
    logit_gemm_kernel<3, 4096>  <<<24,  256, 0, stream>>>(Wd3, ws, OFF_CH3, SZ_L3, 8);
    logit_gemm_kernel<3, 16384> <<<96,  256, 0, stream>>>(Wd2, ws, OFF_CH2, SZ_L2, 32);
    logit_gemm_kernel<3, 65536> <<<384, 256, 0, stream>>>(Wd1, ws, OFF_CH1, SZ_L1, 128);

    logit_reduce_kernel<<<5, 256, 0, stream>>>(ws);
    gate_kernel        <<<1, 128, 0, stream>>>(bca, bd3, bd2, bd1, ws);
    idwt_kernel        <<<512, 256, 0, stream>>>(ws, out);
}